// SelfAttention_74577812128623
// MI455X (gfx1250) — compile-verified
//
#include <hip/hip_runtime.h>
#include <math.h>

typedef float v2f __attribute__((ext_vector_type(2)));
typedef float v8f __attribute__((ext_vector_type(8)));

#define NTOK 4096
#define CH   64
#define NB   4
#define KT   64          // keys per online-softmax pass (4 x 16-wide WMMA tiles)
#define PSTR 66          // LDS row stride for the 16x64 P tile (even -> 8B aligned)

// ---------------------------------------------------------------------------
// Stage 1: QKV projections.
//   Qn[b][n][d] = Wq[d,:]·x[b,:,n] + bq[d]      (layout [B][N][64])
//   Kn[b][n][d] = Wk[d,:]·x[b,:,n] + bk[d]      (layout [B][N][64])
//   Vt[b][e][n] = Wv[e,:]·x[b,:,n] + bv[e]      (layout [B][64][N])
// ---------------------------------------------------------------------------
__global__ __launch_bounds__(256) void proj_kernel(
    const float* __restrict__ x,
    const float* __restrict__ Wq, const float* __restrict__ bq,
    const float* __restrict__ Wk, const float* __restrict__ bk,
    const float* __restrict__ Wv, const float* __restrict__ bv,
    float* __restrict__ Qn, float* __restrict__ Kn, float* __restrict__ Vt)
{
    const int b     = blockIdx.z;
    const int which = blockIdx.y;
    const int tid   = threadIdx.x;
    const float* xb = x + (size_t)b * CH * NTOK;

    if (which < 2) {
        // Q or K: block = 64 d x 4 n ; writes coalesced in d.
        const int d  = tid & 63;
        const int nl = tid >> 6;
        const int n  = blockIdx.x * 4 + nl;
        const float* W    = (which == 0) ? Wq : Wk;
        const float* bias = (which == 0) ? bq : bk;
        float acc = bias[d];
        #pragma unroll 8
        for (int c = 0; c < CH; ++c)
            acc += W[d * CH + c] * xb[(size_t)c * NTOK + n];
        float* out = (which == 0) ? Qn : Kn;
        out[((size_t)b * NTOK + n) * CH + d] = acc;
    } else {
        // V: block = 4 e x 64 n ; x reads and Vt writes coalesced in n.
        const int et = blockIdx.x >> 6;   // 0..15
        const int nt = blockIdx.x & 63;   // 0..63
        const int e  = et * 4 + (tid >> 6);
        const int n  = nt * 64 + (tid & 63);
        float acc = bv[e];
        #pragma unroll 8
        for (int c = 0; c < CH; ++c)
            acc += Wv[e * CH + c] * xb[(size_t)c * NTOK + n];
        Vt[((size_t)b * CH + e) * NTOK + n] = acc;
    }
}

// ---------------------------------------------------------------------------
// Stage 2: flash attention + fused output projection + sigmoid.
// One wave32 per 16-query tile; 64 keys per online-softmax pass.
// fp32 WMMA (16x16x4) throughout; 128 WMMA per pass.
// ---------------------------------------------------------------------------
__global__ __launch_bounds__(256) void flash_kernel(
    const float* __restrict__ Qn, const float* __restrict__ Kn,
    const float* __restrict__ Vt, const float* __restrict__ Wo,
    const float* __restrict__ bo, float* __restrict__ y)
{
    __shared__ float ldsP[8][16 * PSTR];   // per-wave 16x64 P tile

    const int lane = threadIdx.x & 31;
    const int wid  = threadIdx.x >> 5;
    const int wgl  = blockIdx.x * 8 + wid;   // global wave id
    const int b    = wgl >> 8;               // 256 q-tiles per batch
    const int n0   = (wgl & 255) * 16;

    const int lo = lane & 15;
    const int hi = lane >> 4;

    const float* Qb = Qn + (size_t)b * NTOK * CH;
    const float* Kb = Kn + (size_t)b * NTOK * CH;
    const float* Vb = Vt + (size_t)b * CH * NTOK;

    // Q A-fragments (16x64), softmax scale 1/sqrt(64) folded in.
    v2f qa[16];
    #pragma unroll
    for (int kk = 0; kk < 16; ++kk) {
        v2f t = *(const v2f*)(Qb + (size_t)(n0 + lo) * CH + 4 * kk + 2 * hi);
        qa[kk] = t * 0.125f;
    }

    v8f zero8 = {};
    v8f acc[4];                        // out tile 16x64 (4 col-chunks of 16)
    #pragma unroll
    for (int c = 0; c < 4; ++c) acc[c] = zero8;

    float rmax[8], rsum[8];            // per-row softmax state
    #pragma unroll
    for (int i = 0; i < 8; ++i) { rmax[i] = -__builtin_inff(); rsum[i] = 0.0f; }

    float* myP = &ldsP[wid][0];

    for (int t = 0; t < NTOK / KT; ++t) {
        const int k0 = t * KT;

        // ---- S = (Q*scale) . K^T : 4 independent 16-deep WMMA chains ----
        v8f s[4];
        #pragma unroll
        for (int nt = 0; nt < 4; ++nt) s[nt] = zero8;
        #pragma unroll
        for (int kk = 0; kk < 16; ++kk) {
            #pragma unroll
            for (int nt = 0; nt < 4; ++nt) {
                v2f kb = *(const v2f*)(Kb + (size_t)(k0 + nt * 16 + lo) * CH
                                          + 4 * kk + 2 * hi);
                s[nt] = __builtin_amdgcn_wmma_f32_16x16x4_f32(
                            false, qa[kk], false, kb, (short)0, s[nt], false, false);
            }
        }

        // ---- online softmax: one shuffle-reduce set per 64 keys ----
        #pragma unroll
        for (int i = 0; i < 8; ++i) {
            float r = fmaxf(fmaxf(s[0][i], s[1][i]), fmaxf(s[2][i], s[3][i]));
            r = fmaxf(r, __shfl_xor(r, 1, 32));
            r = fmaxf(r, __shfl_xor(r, 2, 32));
            r = fmaxf(r, __shfl_xor(r, 4, 32));
            r = fmaxf(r, __shfl_xor(r, 8, 32));
            const float mnew  = fmaxf(rmax[i], r);
            const float al    = __expf(rmax[i] - mnew);
            rmax[i] = mnew;

            float ps = 0.0f;
            #pragma unroll
            for (int nt = 0; nt < 4; ++nt) {
                const float pv = __expf(s[nt][i] - mnew);
                ps += pv;
                myP[(i + 8 * hi) * PSTR + nt * 16 + lo] = pv;
            }
            ps += __shfl_xor(ps, 1, 32);
            ps += __shfl_xor(ps, 2, 32);
            ps += __shfl_xor(ps, 4, 32);
            ps += __shfl_xor(ps, 8, 32);
            rsum[i] = rsum[i] * al + ps;

            #pragma unroll
            for (int c = 0; c < 4; ++c) acc[c][i] *= al;
        }

        // DS ops are in-order within a wave; wait for P stores to land.
        asm volatile("s_wait_dscnt 0x0" ::: "memory");

        // P A-fragments (16x64) from LDS.
        v2f pa[16];
        #pragma unroll
        for (int kk = 0; kk < 16; ++kk)
            pa[kk] = *(const v2f*)(myP + lo * PSTR + 4 * kk + 2 * hi);

        // ---- acc += P (16x64) x V (64x64) : 4 independent chains ----
        #pragma unroll
        for (int kk = 0; kk < 16; ++kk) {
            #pragma unroll
            for (int c = 0; c < 4; ++c) {
                v2f vb = *(const v2f*)(Vb + (size_t)(c * 16 + lo) * NTOK
                                          + k0 + 4 * kk + 2 * hi);
                acc[c] = __builtin_amdgcn_wmma_f32_16x16x4_f32(
                             false, pa[kk], false, vb, (short)0, acc[c], false, false);
            }
        }
    }

    // ---- normalize rows, fuse y = sigmoid(Wo . out_row + bo) ----
    float part[8];
    #pragma unroll
    for (int i = 0; i < 8; ++i) {
        const float inv = 1.0f / rsum[i];
        float pp = 0.0f;
        #pragma unroll
        for (int c = 0; c < 4; ++c)
            pp += (acc[c][i] * inv) * Wo[c * 16 + lo];
        pp += __shfl_xor(pp, 1, 32);
        pp += __shfl_xor(pp, 2, 32);
        pp += __shfl_xor(pp, 4, 32);
        pp += __shfl_xor(pp, 8, 32);
        part[i] = pp;
    }
    if (lo == 0) {
        const float bov = bo[0];
        #pragma unroll
        for (int i = 0; i < 8; ++i) {
            const int row = i + 8 * hi;
            const float v = part[i] + bov;
            y[(size_t)b * NTOK + n0 + row] = 1.0f / (1.0f + __expf(-v));
        }
    }
}

// ---------------------------------------------------------------------------
extern "C" void kernel_launch(void* const* d_in, const int* in_sizes, int n_in,
                              void* d_out, int out_size, void* d_ws, size_t ws_size,
                              hipStream_t stream)
{
    const float* x  = (const float*)d_in[0];
    const float* Wq = (const float*)d_in[1];
    const float* bq = (const float*)d_in[2];
    const float* Wk = (const float*)d_in[3];
    const float* bk = (const float*)d_in[4];
    const float* Wv = (const float*)d_in[5];
    const float* bv = (const float*)d_in[6];
    const float* Wo = (const float*)d_in[7];
    const float* bo = (const float*)d_in[8];
    float* y = (float*)d_out;

    float* Qn = (float*)d_ws;                         // [B][N][64]  4 MB
    float* Kn = Qn + (size_t)NB * NTOK * CH;          // [B][N][64]  4 MB
    float* Vt = Kn + (size_t)NB * NTOK * CH;          // [B][64][N]  4 MB

    dim3 g1(1024, 3, NB);
    proj_kernel<<<g1, 256, 0, stream>>>(x, Wq, bq, Wk, bk, Wv, bv, Qn, Kn, Vt);

    // B * (N/16) = 1024 wave-tiles, 8 waves (256 threads) per block.
    flash_kernel<<<128, 256, 0, stream>>>(Qn, Kn, Vt, Wo, bo, y);
}